// AdaptiveCrossEntropyLoss_40742059770077
// MI455X (gfx1250) — compile-verified
//
#include <hip/hip_runtime.h>
#include <hip/hip_bf16.h>

// ---------------------------------------------------------------------------
// AdaptiveCrossEntropyLoss for MI455X (gfx1250, wave32, WMMA + async LDS copy)
//
// Shapes: x[4096,1024] f32, target[4096] i32, W[49152,1024] f32, RW[4,1024] f32
// Output (floats): [loss(1), used(1), per_token_nll(4096), closest(4096)]
//
// Dominant cost: x @ W^T = 412 GFLOP fp32 -> V_WMMA_F32_16X16X4_F32 (exact).
// Staging: GLOBAL_LOAD_ASYNC_TO_LDS_B128 + double-buffered LDS (ASYNCcnt).
// 32 tokens per block (dual accumulators) halve L2 traffic for W and raise
// the wmma:ds ratio; each staged B tile feeds two WMMAs.
// ---------------------------------------------------------------------------

typedef __attribute__((ext_vector_type(2))) float v2f;
typedef __attribute__((ext_vector_type(8))) float v8f;

#define DIM      1024
#define NTOK     4096
#define VTOT     49152
#define SLICE_W  2048          // one vocab slice per blockIdx.y (divides cutoffs)
#define NSLICES  24            // 49152 / 2048
#define MT       32            // tokens per block (2 x 16-row WMMA tiles)
#define NMT      128           // 4096 / 32 token blocks
#define KC       32            // K chunk staged in LDS (double buffered)
#define NKC      32            // 1024 / 32
#define KPAD     (KC + 4)      // +4 floats: bank rotation, conflict-free b64
#define TPW      16            // 16-wide column tiles per wave (2048/16/8)

__device__ __forceinline__ float neg_inf() { return -__builtin_inff(); }

// Low 32 bits of a generic pointer to __shared__ = LDS byte address.
__device__ __forceinline__ uint32_t lds_off(const void* p) {
    return (uint32_t)(uintptr_t)p;
}
// Async DMA: 16 bytes global -> LDS per active lane, tracked by ASYNCcnt.
__device__ __forceinline__ void async_ld_b128(uint32_t lds, const float* g) {
    asm volatile("global_load_async_to_lds_b128 %0, %1, off"
                 :: "v"(lds), "v"(g) : "memory");
}
__device__ __forceinline__ void wait_async0() {
    asm volatile("s_wait_asynccnt 0" ::: "memory");
}

// ---------------------------------------------------------------------------
// Kernel 1: router log-softmax. One wave per token.
// ---------------------------------------------------------------------------
__global__ __launch_bounds__(256)
void ace_router(const float* __restrict__ x, const float* __restrict__ RW,
                float* __restrict__ rlogp)
{
    const int w    = threadIdx.x >> 5;
    const int lane = threadIdx.x & 31;
    const int token = blockIdx.x * 8 + w;

    const float* xr = x + (size_t)token * DIM;
    float a0 = 0.f, a1 = 0.f, a2 = 0.f, a3 = 0.f;
    for (int k = lane; k < DIM; k += 32) {
        const float xv = xr[k];
        a0 += xv * RW[0 * DIM + k];
        a1 += xv * RW[1 * DIM + k];
        a2 += xv * RW[2 * DIM + k];
        a3 += xv * RW[3 * DIM + k];
    }
#pragma unroll
    for (int m = 16; m >= 1; m >>= 1) {
        a0 += __shfl_xor(a0, m, 32);
        a1 += __shfl_xor(a1, m, 32);
        a2 += __shfl_xor(a2, m, 32);
        a3 += __shfl_xor(a3, m, 32);
    }
    if (lane == 0) {
        const float mx  = fmaxf(fmaxf(a0, a1), fmaxf(a2, a3));
        const float se  = expf(a0 - mx) + expf(a1 - mx) + expf(a2 - mx) + expf(a3 - mx);
        const float lse = mx + logf(se);
        rlogp[token * 4 + 0] = a0 - lse;
        rlogp[token * 4 + 1] = a1 - lse;
        rlogp[token * 4 + 2] = a2 - lse;
        rlogp[token * 4 + 3] = a3 - lse;
    }
}

// ---------------------------------------------------------------------------
// Kernel 2: logits GEMM (fp32 WMMA, async double-buffered LDS staging)
//           + online softmax partials per slice.
// grid = (128 token blocks, 24 slices), block = 256 (8 waves).
// ---------------------------------------------------------------------------
__global__ __launch_bounds__(256)
void ace_main(const float* __restrict__ x, const int* __restrict__ tgt,
              const float* __restrict__ W,
              float* __restrict__ Pmax, float* __restrict__ Psum,
              int*   __restrict__ Parg, float* __restrict__ Ptgt)
{
    // 62 KB LDS total (under the 64 KB static cap)
    __shared__ float xch[2][2][16][KPAD];      //  9,216 B (2 M-tiles, dbl buf)
    __shared__ float wch[2][8][16][KPAD];      // 36,864 B (dbl buf)
    __shared__ float lscr[8][32][17];          // 17,408 B (logits scratch)
    // cross-wave stat arrays alias lscr (phases separated by barriers)
    float* smaxA = &lscr[0][0][0];             // [8*32]
    float* ssumA = smaxA + 256;                // [8*32]
    float* stgtA = smaxA + 512;                // [8*32]
    int*   sargA = (int*)(smaxA + 768);        // [8*32]

    const int mbase = blockIdx.x * MT;
    const int slice = blockIdx.y;
    const int tid   = threadIdx.x;
    const int w     = tid >> 5;
    const int lane  = tid & 31;
    const int hi    = lane >> 4;     // half-wave: K offset 0 / 2
    const int lq    = lane & 15;     // M (A) / N (B)
    const int slice_lo = slice * SLICE_W;

    const int token = mbase + lane;  // reduction: lane owns one of 32 tokens
    const int tg    = tgt[token];

    // running online-softmax state (one token per lane, all 32 lanes)
    float mrun = neg_inf(), srun = 0.f, trun = 0.f;
    int   argrun = 0;

    for (int ti = 0; ti < TPW; ++ti) {
        const int tile  = w * TPW + ti;              // wave owns contiguous tiles
        const int vbase = slice_lo + tile * 16;      // increasing -> argmax first-hit

        // enqueue async global->LDS copies of x and W chunk `kc` into buffer `b`
        auto issue = [&](int b, int kc) {
            {   // x: 2 tiles x 16 rows x 8 float4 = 256, one per thread
                const int mt = tid >> 7, r = (tid >> 3) & 15, c4 = tid & 7;
                async_ld_b128(lds_off(&xch[b][mt][r][c4 * 4]),
                              x + (size_t)(mbase + mt * 16 + r) * DIM
                                + kc * KC + c4 * 4);
            }
#pragma unroll
            for (int i = 0; i < 4; ++i) {            // W: 16 rows x 8 float4
                const int f = i * 32 + lane;
                const int r = f >> 3, c4 = f & 7;
                async_ld_b128(lds_off(&wch[b][w][r][c4 * 4]),
                              W + (size_t)(vbase + r) * DIM + kc * KC + c4 * 4);
            }
        };

        v8f c0 = {0.f, 0.f, 0.f, 0.f, 0.f, 0.f, 0.f, 0.f};
        v8f c1 = {0.f, 0.f, 0.f, 0.f, 0.f, 0.f, 0.f, 0.f};
        issue(0, 0);                                 // prologue fill of buffer 0

        for (int kc = 0; kc < NKC; ++kc) {
            const int cur = kc & 1;
            wait_async0();                           // my copies into buf[cur] done
            __syncthreads();                         // everyone's copies visible
            if (kc + 1 < NKC) issue(cur ^ 1, kc + 1);// overlap next copy w/ compute

            // 16 x V_WMMA_F32_16X16X4_F32, B shared by two accumulators
#pragma unroll
            for (int k4 = 0; k4 < 8; ++k4) {
                const int koff = k4 * 4 + 2 * hi;
                const v2f a0 = *(const v2f*)&xch[cur][0][lq][koff];
                const v2f a1 = *(const v2f*)&xch[cur][1][lq][koff];
                const v2f bb = *(const v2f*)&wch[cur][w][lq][koff];
                c0 = __builtin_amdgcn_wmma_f32_16x16x4_f32(
                         false, a0, false, bb, (short)0, c0, false, false);
                c1 = __builtin_amdgcn_wmma_f32_16x16x4_f32(
                         false, a1, false, bb, (short)0, c1, false, false);
            }
        }

        // dump both C tiles: lane holds rows (r + 8*hi), column lq
#pragma unroll
        for (int r = 0; r < 8; ++r) {
            lscr[w][r + 8 * hi][lq]      = c0[r];
            lscr[w][16 + r + 8 * hi][lq] = c1[r];
        }
        __builtin_amdgcn_wave_barrier();

        // per-token tile reduction + online merge: row = lane (all 32 lanes busy,
        // 17-float row stride -> conflict-free)
        {
            float tmax = neg_inf(); int tn = 0;
#pragma unroll
            for (int n = 0; n < 16; ++n) {
                const float v = lscr[w][lane][n];
                if (v > tmax) { tmax = v; tn = n; }    // strict > keeps first max
            }
            float ts = 0.f;
#pragma unroll
            for (int n = 0; n < 16; ++n) ts += __expf(lscr[w][lane][n] - tmax);
            if (tg >= vbase && tg < vbase + 16) trun = lscr[w][lane][tg - vbase];
            if (tmax > mrun) {
                srun = srun * __expf(mrun - tmax) + ts;
                mrun = tmax; argrun = vbase + tn;
            } else {
                srun += ts * __expf(tmax - mrun);      // equal max keeps earlier idx
            }
        }
        __syncthreads();   // buffers + lscr reusable for next ti
    }

    // merge the 8 waves' partials (waves own increasing vocab ranges)
    smaxA[w * 32 + lane] = mrun;
    ssumA[w * 32 + lane] = srun;
    stgtA[w * 32 + lane] = trun;
    sargA[w * 32 + lane] = argrun;
    __syncthreads();
    if (tid < 32) {
        const int t = tid;
        float M = smaxA[t], S = ssumA[t]; int A = sargA[t];
#pragma unroll
        for (int q = 1; q < 8; ++q) {
            const float m2 = smaxA[q * 32 + t], s2 = ssumA[q * 32 + t];
            const int   a2 = sargA[q * 32 + t];
            if (m2 > M) { S = S * __expf(M - m2) + s2; M = m2; A = a2; }
            else        { S += s2 * __expf(m2 - M); }   // tie keeps lower vocab idx
        }
        const int tg2 = tgt[mbase + t];
        float TG = 0.f;
        if (tg2 >= slice_lo && tg2 < slice_lo + SLICE_W)
            TG = stgtA[((tg2 - slice_lo) >> 8) * 32 + t];   // 256 cols per wave
        const size_t p = ((size_t)slice * NMT + blockIdx.x) * MT + t;
        Pmax[p] = M; Psum[p] = S; Parg[p] = A; Ptgt[p] = TG;
    }
}

// ---------------------------------------------------------------------------
// Kernel 3: per-token merge over its cluster's slices -> nll + closest.
// ---------------------------------------------------------------------------
__global__ __launch_bounds__(256)
void ace_token(const int* __restrict__ tgt,
               const float* __restrict__ Pmax, const float* __restrict__ Psum,
               const int* __restrict__ Parg, const float* __restrict__ Ptgt,
               const float* __restrict__ rlogp, float* __restrict__ out)
{
    const int t = blockIdx.x * 256 + threadIdx.x;
    if (t >= NTOK) return;
    const int tg = tgt[t];

    int c, lo, hiv;
    if      (tg < 16384) { c = 0; lo = 0;     hiv = 16384; }
    else if (tg < 32768) { c = 1; lo = 16384; hiv = 32768; }
    else if (tg < 40960) { c = 2; lo = 32768; hiv = 40960; }
    else                 { c = 3; lo = 40960; hiv = 49152; }

    const int s0 = lo >> 11, s1 = hiv >> 11;   // slices of width 2048
    const int mtile = t >> 5, tt = t & 31;     // 32-token blocks

    float M = neg_inf(), S = 0.f; int A = 0;
    for (int s = s0; s < s1; ++s) {
        const size_t p = ((size_t)s * NMT + mtile) * MT + tt;
        const float m2 = Pmax[p], s2 = Psum[p]; const int a2 = Parg[p];
        if (m2 > M) { S = S * expf(M - m2) + s2; M = m2; A = a2; }
        else        { S += s2 * expf(m2 - M); }
    }
    const size_t ptg = ((size_t)(tg >> 11) * NMT + mtile) * MT + tt;
    const float tlog = Ptgt[ptg];
    const float lse  = M + logf(S);
    const float nll  = (lse - tlog) - rlogp[t * 4 + c];

    out[2 + t]        = nll;          // per-token nll
    out[2 + NTOK + t] = (float)A;     // closest (global vocab index = local + lo)
}

// ---------------------------------------------------------------------------
// Kernel 4: deterministic reduction -> loss, used.
// ---------------------------------------------------------------------------
__global__ __launch_bounds__(256)
void ace_final(const int* __restrict__ tgt, float* __restrict__ out)
{
    __shared__ float rsum[256];
    __shared__ int   rcnt[256];
    const int tid = threadIdx.x;
    float s = 0.f; int cnt = 0;
    for (int i = tid; i < NTOK; i += 256) {
        s += out[2 + i];
        const int tg = tgt[i];
        cnt += (tg >= 0 && tg < VTOT) ? 1 : 0;
    }
    rsum[tid] = s; rcnt[tid] = cnt;
    __syncthreads();
    for (int st = 128; st >= 1; st >>= 1) {
        if (tid < st) { rsum[tid] += rsum[tid + st]; rcnt[tid] += rcnt[tid + st]; }
        __syncthreads();
    }
    if (tid == 0) {
        const int used = rcnt[0];
        const int d    = used < 1 ? 1 : used;
        out[0] = rsum[0] / (float)d;
        out[1] = (float)used;
    }
}

// ---------------------------------------------------------------------------
extern "C" void kernel_launch(void* const* d_in, const int* in_sizes, int n_in,
                              void* d_out, int out_size, void* d_ws, size_t ws_size,
                              hipStream_t stream)
{
    const float* x  = (const float*)d_in[0];   // [4096,1024]
    const int*   tg = (const int*)d_in[1];     // [4096]
    const float* W  = (const float*)d_in[2];   // [49152,1024]
    const float* RW = (const float*)d_in[3];   // [4,1024]
    float* out = (float*)d_out;                // 8194 floats

    // workspace layout (floats): rlogp | Pmax | Psum | Ptgt | Parg(int)
    const size_t NP = (size_t)NSLICES * NMT * MT;   // 98304
    float* rlogp = (float*)d_ws;
    float* Pmax  = rlogp + (size_t)NTOK * 4;
    float* Psum  = Pmax + NP;
    float* Ptgt  = Psum + NP;
    int*   Parg  = (int*)(Ptgt + NP);
    (void)in_sizes; (void)n_in; (void)out_size; (void)ws_size;

    ace_router<<<NTOK / 8, 256, 0, stream>>>(x, RW, rlogp);

    dim3 g2(NMT, NSLICES);
    ace_main<<<g2, 256, 0, stream>>>(x, tg, W, Pmax, Psum, Parg, Ptgt);

    ace_token<<<NTOK / 256, 256, 0, stream>>>(tg, Pmax, Psum, Parg, Ptgt, rlogp, out);

    ace_final<<<1, 256, 0, stream>>>(tg, out);
}